// HyperKA_60172491817238
// MI455X (gfx1250) — compile-verified
//
#include <hip/hip_runtime.h>
#include <cmath>

#define EPS       1e-5f
#define MAX_NORM  (1.0f - 1e-5f)
#define DIM       64
#define N_LAYERS  2

typedef __attribute__((ext_vector_type(2))) float v2f;
typedef __attribute__((ext_vector_type(8))) float v8f;

// ---- wave32 reduction: sum across all 32 lanes --------------------------
__device__ __forceinline__ float wave_sum(float v) {
#pragma unroll
    for (int off = 16; off > 0; off >>= 1)
        v += __shfl_xor(v, off, 32);
    return v;
}

// ---- kernel 1: x = proj(exp_map_zero(emb)); also stash x0 ---------------
__global__ void __launch_bounds__(256)
hk_init(const float* __restrict__ emb, float* __restrict__ x,
        float* __restrict__ x0, int N) {
    int gtid = blockIdx.x * blockDim.x + threadIdx.x;
    int row  = gtid >> 5;
    int lane = gtid & 31;
    if (row >= N) return;
    float2 v = ((const float2*)(emb + (size_t)row * DIM))[lane];
    float  s    = wave_sum(v.x * v.x + v.y * v.y);
    float  rawn = sqrtf(s);
    float  n    = fmaxf(rawn, EPS);
    float  sc   = tanhf(n) / n;
    float2 y;  y.x = v.x * sc;  y.y = v.y * sc;
    float  ny = fmaxf(sc * rawn, EPS);
    if (ny > MAX_NORM) { float p = MAX_NORM / ny; y.x *= p; y.y *= p; }
    ((float2*)(x  + (size_t)row * DIM))[lane] = y;
    ((float2*)(x0 + (size_t)row * DIM))[lane] = y;
}

// ---- kernel 2: h = log_map_zero(x) @ W  (fp32 WMMA) ---------------------
// Block = 256 threads = 8 waves; each wave owns 16 rows -> 128 rows/block.
// Tangent tile staged in LDS (stride 65 to kill bank conflicts), W in LDS.
__global__ void __launch_bounds__(256)
hk_tangent_gemm(const float* __restrict__ x, const float* __restrict__ W,
                float* __restrict__ h, int N) {
    __shared__ float Tsh[128 * 65];
    __shared__ float Wsh[64 * 64];
    const int tid  = threadIdx.x;
    const int wave = tid >> 5;
    const int lane = tid & 31;
    const int rowBase = blockIdx.x * 128;

    // cooperative load of W (64x64 fp32, row-major)
#pragma unroll
    for (int i = 0; i < 16; ++i) Wsh[tid + i * 256] = W[tid + i * 256];

    // tangent = arctanh(clip(|x|)) * x / clip(|x|), one row per wave-iter
    for (int r = 0; r < 16; ++r) {
        int lr = wave * 16 + r;
        int g  = rowBase + lr;
        float2 v; v.x = 0.f; v.y = 0.f;
        if (g < N) v = ((const float2*)(x + (size_t)g * DIM))[lane];
        float s  = wave_sum(v.x * v.x + v.y * v.y);
        float n  = fminf(fmaxf(sqrtf(s), EPS), MAX_NORM);
        float at = 0.5f * logf((1.f + n) / (1.f - n));   // arctanh(n)
        float sc = at / n;
        Tsh[lr * 65 + 2 * lane]     = v.x * sc;
        Tsh[lr * 65 + 2 * lane + 1] = v.y * sc;
    }
    __syncthreads();

    // V_WMMA_F32_16X16X4_F32 lane layout (ISA 7.12.2):
    //  A 16x4 : lanes 0-15 -> M=lane, K={0,1}; lanes 16-31 -> M=lane-16, K={2,3}
    //  B 4x16 : lanes 0-15 -> N=lane, K={0,1}; lanes 16-31 -> N=lane-16, K={2,3}
    //  C 16x16: vgpr r, lanes 0-15 -> M=r; lanes 16-31 -> M=r+8; N=lane&15
    const int m     = lane & 15;
    const int khalf = lane >> 4;
#pragma unroll
    for (int nt = 0; nt < 4; ++nt) {
        const int n = nt * 16 + m;
        v8f c = {0.f, 0.f, 0.f, 0.f, 0.f, 0.f, 0.f, 0.f};
#pragma unroll
        for (int kk = 0; kk < 16; ++kk) {
            const int kb = kk * 4 + khalf * 2;
            v2f a = { Tsh[(wave * 16 + m) * 65 + kb],
                      Tsh[(wave * 16 + m) * 65 + kb + 1] };
            v2f b = { Wsh[kb * 64 + n],
                      Wsh[(kb + 1) * 64 + n] };
            c = __builtin_amdgcn_wmma_f32_16x16x4_f32(
                    false, a, false, b, (short)0, c, false, false);
        }
#pragma unroll
        for (int r = 0; r < 8; ++r) {
            int g = rowBase + wave * 16 + r + 8 * khalf;
            if (g < N) h[(size_t)g * DIM + n] = c[r];
        }
    }
}

// ---- kernel 3: agg[row] += val * h[col]  (one wave per edge) ------------
__global__ void __launch_bounds__(256)
hk_spmm(const float* __restrict__ h, const int* __restrict__ er,
        const int* __restrict__ ec, const float* __restrict__ ev,
        float* __restrict__ agg, int E) {
    int gtid = blockIdx.x * blockDim.x + threadIdx.x;
    int e    = gtid >> 5;
    int lane = gtid & 31;
    if (e >= E) return;
    int   r  = er[e];
    int   ci = ec[e];
    float v  = ev[e];
    float2 m = ((const float2*)(h + (size_t)ci * DIM))[lane];
    float* dst = agg + (size_t)r * DIM + 2 * lane;
    unsafeAtomicAdd(dst,     v * m.x);   // global_atomic_add_f32
    unsafeAtomicAdd(dst + 1, v * m.y);
}

// ---- kernel 4: x = proj(exp_map_zero(tanh(agg))) ------------------------
__global__ void __launch_bounds__(256)
hk_post(const float* __restrict__ agg, float* __restrict__ x, int N) {
    int gtid = blockIdx.x * blockDim.x + threadIdx.x;
    int row  = gtid >> 5;
    int lane = gtid & 31;
    if (row >= N) return;
    float2 a = ((const float2*)(agg + (size_t)row * DIM))[lane];
    float2 t;  t.x = tanhf(a.x);  t.y = tanhf(a.y);
    float  s    = wave_sum(t.x * t.x + t.y * t.y);
    float  rawn = sqrtf(s);
    float  n    = fmaxf(rawn, EPS);
    float  sc   = tanhf(n) / n;
    float2 y;  y.x = t.x * sc;  y.y = t.y * sc;
    float  ny = fmaxf(sc * rawn, EPS);
    if (ny > MAX_NORM) { float p = MAX_NORM / ny; y.x *= p; y.y *= p; }
    ((float2*)(x + (size_t)row * DIM))[lane] = y;
}

// ---- kernel 5: out = proj(mobius_add(x, x0)); x0 lives in `out` ---------
__global__ void __launch_bounds__(256)
hk_mobius(const float* __restrict__ x, float* __restrict__ out, int N) {
    int gtid = blockIdx.x * blockDim.x + threadIdx.x;
    int row  = gtid >> 5;
    int lane = gtid & 31;
    if (row >= N) return;
    float2 a = ((const float2*)(x   + (size_t)row * DIM))[lane];  // x
    float2 b = ((const float2*)(out + (size_t)row * DIM))[lane];  // x0
    float x2 = wave_sum(a.x * a.x + a.y * a.y);
    float y2 = wave_sum(b.x * b.x + b.y * b.y);
    float xy = wave_sum(a.x * b.x + a.y * b.y);
    float ca  = 1.f + 2.f * xy + y2;
    float cb  = 1.f - x2;
    float den = fmaxf(1.f + 2.f * xy + x2 * y2, EPS);
    float2 o;
    o.x = (ca * a.x + cb * b.x) / den;
    o.y = (ca * a.y + cb * b.y) / den;
    float s  = wave_sum(o.x * o.x + o.y * o.y);
    float on = fmaxf(sqrtf(s), EPS);
    if (on > MAX_NORM) { float p = MAX_NORM / on; o.x *= p; o.y *= p; }
    ((float2*)(out + (size_t)row * DIM))[lane] = o;
}

extern "C" void kernel_launch(void* const* d_in, const int* in_sizes, int n_in,
                              void* d_out, int out_size, void* d_ws, size_t ws_size,
                              hipStream_t stream) {
    const float* emb   = (const float*)d_in[0];
    const float* Wall  = (const float*)d_in[1];
    const int*   erows = (const int*)d_in[2];
    const int*   ecols = (const int*)d_in[3];
    const float* evals = (const float*)d_in[4];

    const int N = in_sizes[0] / DIM;
    const int E = in_sizes[2];

    float* out = (float*)d_out;
    float* x   = (float*)d_ws;                 // N*64 fp32
    float* h   = x + (size_t)N * DIM;          // N*64 fp32
    float* agg = h + (size_t)N * DIM;          // N*64 fp32

    dim3 blk(256);
    const int rowBlocks  = (N + 7) / 8;        // 8 waves/block, 1 row/wave
    const int gemmBlocks = (N + 127) / 128;    // 128 rows/block
    const int spmmBlocks = (E + 7) / 8;        // 8 edges/block

    hk_init<<<rowBlocks, blk, 0, stream>>>(emb, x, out, N);
    for (int l = 0; l < N_LAYERS; ++l) {
        hk_tangent_gemm<<<gemmBlocks, blk, 0, stream>>>(x, Wall + (size_t)l * DIM * DIM, h, N);
        hipMemsetAsync(agg, 0, (size_t)N * DIM * sizeof(float), stream);
        hk_spmm<<<spmmBlocks, blk, 0, stream>>>(h, erows, ecols, evals, agg, E);
        hk_post<<<rowBlocks, blk, 0, stream>>>(agg, x, N);
    }
    hk_mobius<<<rowBlocks, blk, 0, stream>>>(x, out, N);
}